// SimpleGraphConv_12317966205223
// MI455X (gfx1250) — compile-verified
//
#include <hip/hip_runtime.h>
#include <stdint.h>

#define N_NODES 8192
#define DIM 256
#define CAP 4096   // max neighbors compacted per node (mean degree ~33; Poisson tail => safe)

typedef __attribute__((ext_vector_type(2))) float v2f;
typedef __attribute__((ext_vector_type(8))) float v8f;

// ---------------------------------------------------------------------------
// 1) Init adjacency bitmap: zero everything, set diagonal bit (self loops).
//    bitmap is N rows x 256 u32 words (8192 bits per row).
// ---------------------------------------------------------------------------
__global__ void gat_init_bitmap(uint32_t* __restrict__ bitmap) {
    int w = blockIdx.x * 256 + threadIdx.x;         // 0 .. N*256-1
    int row = w >> 8;
    int ww  = w & 255;
    uint32_t v = (ww == (row >> 5)) ? (1u << (row & 31)) : 0u;
    bitmap[w] = v;
}

// ---------------------------------------------------------------------------
// 2) Scatter edges into the bitmap (dedupes duplicate edges automatically).
//    edge_index flat: [0..E) = src row, [E..2E) = dst row.
// ---------------------------------------------------------------------------
__global__ void gat_build_adj(const int* __restrict__ ei, uint32_t* __restrict__ bitmap, int E) {
    int e = blockIdx.x * 256 + threadIdx.x;
    if (e >= E) return;
    int src = ei[e];
    int dst = ei[E + e];
    atomicOr(&bitmap[(size_t)src * 256 + (dst >> 5)], 1u << (dst & 31));
}

// ---------------------------------------------------------------------------
// 3) h = x @ W + b via V_WMMA_F32_16X16X4_F32.
//    One wave per 16x32 output tile (two accumulators share the A fragment).
//    A frag  (16x4):  lane L -> row L&15, a[v] = x[row][k0 + v + 2*(L>>4)]
//    B frag  (4x16):  lane L -> col L&15, b[v] = W[k0 + v + 2*(L>>4)][col]
//    C/D     (16x16): lane L -> col L&15, c[v] = h[row0 + v + 8*(L>>4)][col]
// ---------------------------------------------------------------------------
__global__ void gat_linear_wmma(const float* __restrict__ x,
                                const float* __restrict__ W,
                                const float* __restrict__ bias,
                                float* __restrict__ h) {
    const int lane = threadIdx.x;        // 0..31, one wave per block
    const int hi   = lane >> 4;          // 0 or 1
    const int lo   = lane & 15;
    const int row0 = blockIdx.x * 16;    // M tile
    const int col0 = blockIdx.y * 32;    // N tile (two 16-wide subtiles)

    v8f c0 = {};
    v8f c1 = {};

    const float* xrow = x + (size_t)(row0 + lo) * DIM;

    #pragma unroll 4
    for (int k0 = 0; k0 < DIM; k0 += 4) {
        v2f a;
        const float* xp = xrow + k0 + 2 * hi;
        a.x = xp[0];
        a.y = xp[1];

        const float* wp = W + (size_t)(k0 + 2 * hi) * DIM + lo;
        v2f b0, b1;
        b0.x = wp[col0];          b0.y = wp[col0 + DIM];
        b1.x = wp[col0 + 16];     b1.y = wp[col0 + 16 + DIM];

        c0 = __builtin_amdgcn_wmma_f32_16x16x4_f32(false, a, false, b0,
                                                   (short)0, c0, false, false);
        c1 = __builtin_amdgcn_wmma_f32_16x16x4_f32(false, a, false, b1,
                                                   (short)0, c1, false, false);
    }

    const float bv0 = bias[col0 + lo];
    const float bv1 = bias[col0 + 16 + lo];
    #pragma unroll
    for (int v = 0; v < 8; ++v) {
        int row = row0 + v + 8 * hi;
        h[(size_t)row * DIM + col0 + lo]      = c0[v] + bv0;
        h[(size_t)row * DIM + col0 + 16 + lo] = c1[v] + bv1;
    }
}

// ---------------------------------------------------------------------------
// 4) s_i[n] = h[n] . a_w[0:256],  s_j[n] = h[n] . a_w[256:512]
//    One wave per node, shuffle reduction (wave32).
// ---------------------------------------------------------------------------
__global__ void gat_scores(const float* __restrict__ h,
                           const float* __restrict__ a_w,
                           float* __restrict__ si, float* __restrict__ sj) {
    int node = blockIdx.x * 8 + (threadIdx.x >> 5);
    int lane = threadIdx.x & 31;
    const float* hp = h + (size_t)node * DIM;
    float ai = 0.f, aj = 0.f;
    #pragma unroll
    for (int t = lane; t < DIM; t += 32) {
        float v = hp[t];
        ai += v * a_w[t];
        aj += v * a_w[DIM + t];
    }
    #pragma unroll
    for (int off = 16; off > 0; off >>= 1) {
        ai += __shfl_down(ai, off, 32);
        aj += __shfl_down(aj, off, 32);
    }
    if (lane == 0) { si[node] = ai; sj[node] = aj; }
}

// ---------------------------------------------------------------------------
// 5) Per-node sparse softmax + aggregation. One 256-thread block per node.
//    Pass A: scan bitmap row (1 word/thread), compact (j, leaky_relu score)
//            pairs into LDS.
//    Pass B: block-reduce max, convert scores -> exp weights, reduce sum.
//    Pass C: thread t owns channel t; accumulate over compact neighbor list.
// ---------------------------------------------------------------------------
__global__ void gat_aggregate(const float* __restrict__ h,
                              const uint32_t* __restrict__ bitmap,
                              const float* __restrict__ si,
                              const float* __restrict__ sj,
                              const float* __restrict__ a_b,
                              float* __restrict__ out) {
    __shared__ uint32_t s_idx[CAP];
    __shared__ float    s_w[CAP];
    __shared__ float    s_red[256];
    __shared__ uint32_t s_cnt;

    const int i   = blockIdx.x;
    const int tid = threadIdx.x;
    if (tid == 0) s_cnt = 0;
    __syncthreads();

    const float sii = si[i];
    const float ab  = a_b[0];

    // Pass A: compaction
    uint32_t bits = bitmap[(size_t)i * 256 + tid];
    while (bits) {
        int b = __ffs(bits) - 1;
        bits &= bits - 1;
        int j = tid * 32 + b;
        float sc = sii + sj[j] + ab;
        sc = (sc > 0.f) ? sc : 0.2f * sc;        // leaky_relu(0.2)
        uint32_t slot = atomicAdd(&s_cnt, 1u);
        if (slot < CAP) { s_idx[slot] = (uint32_t)j; s_w[slot] = sc; }
    }
    __syncthreads();
    uint32_t cnt = s_cnt < CAP ? s_cnt : CAP;     // always >= 1 (self loop)

    // Pass B1: max reduce
    float m = -3.4e38f;
    for (uint32_t e = tid; e < cnt; e += 256) m = fmaxf(m, s_w[e]);
    s_red[tid] = m;
    __syncthreads();
    #pragma unroll
    for (int st = 128; st > 0; st >>= 1) {
        if (tid < st) s_red[tid] = fmaxf(s_red[tid], s_red[tid + st]);
        __syncthreads();
    }
    m = s_red[0];
    __syncthreads();

    // Pass B2: exp + sum reduce (weights written back in place)
    float sum = 0.f;
    for (uint32_t e = tid; e < cnt; e += 256) {
        float w = __expf(s_w[e] - m);
        s_w[e] = w;
        sum += w;
    }
    s_red[tid] = sum;
    __syncthreads();
    #pragma unroll
    for (int st = 128; st > 0; st >>= 1) {
        if (tid < st) s_red[tid] += s_red[tid + st];
        __syncthreads();
    }
    const float inv = 1.0f / s_red[0];

    // Pass C: weighted gather of h rows; thread tid owns output channel tid.
    float acc = 0.f;
    for (uint32_t e = 0; e < cnt; ++e) {
        acc += s_w[e] * h[(size_t)s_idx[e] * DIM + tid];
    }
    out[(size_t)i * DIM + tid] = acc * inv;
}

// ---------------------------------------------------------------------------
// Launcher
// ---------------------------------------------------------------------------
extern "C" void kernel_launch(void* const* d_in, const int* in_sizes, int n_in,
                              void* d_out, int out_size, void* d_ws, size_t ws_size,
                              hipStream_t stream) {
    const float* x   = (const float*)d_in[0];
    const int*   ei  = (const int*)  d_in[1];
    const float* W_w = (const float*)d_in[2];
    const float* W_b = (const float*)d_in[3];
    const float* a_w = (const float*)d_in[4];
    const float* a_b = (const float*)d_in[5];
    float* out = (float*)d_out;

    const int E = in_sizes[1] / 2;

    char* ws = (char*)d_ws;
    float*    h      = (float*)ws;                                 // 8 MB
    uint32_t* bitmap = (uint32_t*)(ws + (size_t)8 * 1024 * 1024);  // 8 MB
    float*    si     = (float*)(ws + (size_t)16 * 1024 * 1024);    // 32 KB
    float*    sj     = si + N_NODES;                               // 32 KB

    // bitmap words = N*256 = 2,097,152 -> 8192 blocks of 256
    hipLaunchKernelGGL(gat_init_bitmap, dim3(N_NODES), dim3(256), 0, stream, bitmap);
    hipLaunchKernelGGL(gat_build_adj, dim3((E + 255) / 256), dim3(256), 0, stream,
                       ei, bitmap, E);
    // 16x32 tiles: grid (8192/16, 256/32), one wave per block
    hipLaunchKernelGGL(gat_linear_wmma, dim3(N_NODES / 16, DIM / 32), dim3(32), 0, stream,
                       x, W_w, W_b, h);
    hipLaunchKernelGGL(gat_scores, dim3(N_NODES / 8), dim3(256), 0, stream,
                       h, a_w, si, sj);
    hipLaunchKernelGGL(gat_aggregate, dim3(N_NODES), dim3(256), 0, stream,
                       h, bitmap, si, sj, a_b, out);
}